// BestEllipseLoss_5720896438361
// MI455X (gfx1250) — compile-verified
//
#include <hip/hip_runtime.h>
#include <stdint.h>
#include <math.h>

// ---------------------------------------------------------------------------
// BestEllipseLoss for MI455X (gfx1250).
// One 256-thread block per image; image staged to LDS (async-to-LDS on CDNA5),
// all analysis passes run out of LDS. Final mean over B losses done with
// chained V_WMMA_F32_16X16X4_F32 row-sum reductions.
// ---------------------------------------------------------------------------

#define H_    128
#define W_    128
#define NPIX  (H_ * W_)
#define NLEV  5
#define NT    256
#define NW    (NT / 32)
#define EPSF  1e-8f

__device__ __constant__ float c_lev[NLEV] = {0.3f, 0.4f, 0.5f, 0.6f, 0.7f};

// ------------------------- staging: global -> LDS ---------------------------

#if __has_builtin(__builtin_amdgcn_global_load_async_to_lds_b128)
#define HAVE_ASYNC_LDS 1
#else
#define HAVE_ASYNC_LDS 0
#endif

typedef int v4i_b128 __attribute__((vector_size(16)));
typedef __attribute__((address_space(1))) v4i_b128* g_v4i_p;   // global (AS1)
typedef __attribute__((address_space(3))) v4i_b128* l_v4i_p;   // LDS (AS3)

__device__ __forceinline__ void stage_image(float* __restrict__ dst_lds,
                                            const float* __restrict__ src) {
#if HAVE_ASYNC_LDS
  for (int i = threadIdx.x * 4; i < NPIX; i += NT * 4) {
    __builtin_amdgcn_global_load_async_to_lds_b128(
        (g_v4i_p)(void*)(src + i), (l_v4i_p)(void*)(dst_lds + i), 0, 0);
  }
#if __has_builtin(__builtin_amdgcn_s_wait_asynccnt)
  __builtin_amdgcn_s_wait_asynccnt(0);
#else
  asm volatile("s_wait_asynccnt 0" ::: "memory");
#endif
#else
  for (int i = threadIdx.x * 4; i < NPIX; i += NT * 4) {
    const float4 v = *reinterpret_cast<const float4*>(src + i);
    *reinterpret_cast<float4*>(dst_lds + i) = v;
  }
#endif
  __syncthreads();
}

// ------------------------- block reductions ---------------------------------

__device__ __forceinline__ void block_minmax(const float* __restrict__ img,
                                             float* sred, float* sout,
                                             float& mn_out, float& mx_out) {
  const int tid = threadIdx.x, lane = tid & 31, wave = tid >> 5;
  float mn = 3.402823466e38f, mx = -3.402823466e38f;
  for (int i = tid; i < NPIX; i += NT) {
    const float v = img[i];
    mn = fminf(mn, v);
    mx = fmaxf(mx, v);
  }
#pragma unroll
  for (int off = 16; off > 0; off >>= 1) {
    mn = fminf(mn, __shfl_xor(mn, off, 32));
    mx = fmaxf(mx, __shfl_xor(mx, off, 32));
  }
  if (lane == 0) { sred[wave * 2] = mn; sred[wave * 2 + 1] = mx; }
  __syncthreads();
  if (tid == 0) {
    float a = sred[0], c = sred[1];
    for (int w = 1; w < NW; ++w) {
      a = fminf(a, sred[w * 2]);
      c = fmaxf(c, sred[w * 2 + 1]);
    }
    sout[0] = a; sout[1] = c;
  }
  __syncthreads();
  mn_out = sout[0];
  mx_out = sout[1];
}

template <int N>
__device__ __forceinline__ void block_sum(const float* acc, float* sred, float* sout) {
  const int lane = threadIdx.x & 31, wave = threadIdx.x >> 5;
#pragma unroll
  for (int i = 0; i < N; ++i) {
    float v = acc[i];
#pragma unroll
    for (int off = 16; off > 0; off >>= 1) v += __shfl_xor(v, off, 32);
    if (lane == 0) sred[wave * 32 + i] = v;
  }
  __syncthreads();
  if (threadIdx.x < N) {
    float s = 0.f;
    for (int w = 0; w < NW; ++w) s += sred[w * 32 + threadIdx.x];
    sout[threadIdx.x] = s;
  }
  __syncthreads();
}

// ---------------------- ellipse fit from raw sums ---------------------------
// s = {Sw, Swx, Swy, Swxx, Swyy, Swxy}; matches reference: m00 = Sw + EPS,
// central moments expanded exactly around cx = Swx/m00 (not Swx/Sw).

__device__ __forceinline__ void fit_params(const float* s, float* p) {
  const float S = s[0];
  const float m00 = S + EPSF;
  const float cx = s[1] / m00, cy = s[2] / m00;
  const float mu20 = (s[3] - 2.f * cx * s[1] + cx * cx * S) / m00;
  const float mu02 = (s[4] - 2.f * cy * s[2] + cy * cy * S) / m00;
  const float mu11 = (s[5] - cx * s[2] - cy * s[1] + cx * cy * S) / m00;
  const float d = mu20 - mu02;
  const float theta = 0.5f * atan2f(2.f * mu11, d);
  const float common = sqrtf(d * d + 4.f * mu11 * mu11);
  const float a = sqrtf(fmaxf(2.f * (mu20 + mu02 + common), EPSF));
  const float b = sqrtf(fmaxf(2.f * (mu20 + mu02 - common), EPSF));
  p[0] = cx; p[1] = cy; p[2] = theta; p[3] = a; p[4] = b;
}

// ------------------------------ main kernel ---------------------------------

__global__ __launch_bounds__(NT) void ellipse_best_loss_kernel(
    const float* __restrict__ outg, const float* __restrict__ tgtg,
    float* __restrict__ loss) {
  __shared__ __align__(16) float img[NPIX];     // 64 KiB staged image
  __shared__ float sred[NW * 32];
  __shared__ float sout[32];
  __shared__ float sparams[NLEV * 8];           // cx,cy,theta,a,b,cos,sin per level
  __shared__ float sbestp[5];
  __shared__ int   sbesti;

  const int tid = threadIdx.x;
  const int blk = blockIdx.x;

  // ===================== phase 1: target image ==========================
  stage_image(img, tgtg + (size_t)blk * NPIX);

  float mn, mx;
  block_minmax(img, sred, sout, mn, mx);
  const float inv  = 1.0f / (mx - mn + EPSF);
  const float tmax = (mx - mn) * inv;           // max of normalized image

  float thr[NLEV];
#pragma unroll
  for (int l = 0; l < NLEV; ++l) thr[l] = c_lev[l] * tmax;

  // --- 5-level raw moments (write normalized value back to LDS) ---
  float acc[6 * NLEV];
#pragma unroll
  for (int k = 0; k < 6 * NLEV; ++k) acc[k] = 0.f;

  for (int i = tid; i < NPIX; i += NT) {
    const float t = (img[i] - mn) * inv;
    img[i] = t;                                  // exclusive per-thread slot
    const float x = (float)(i & (W_ - 1));
    const float y = (float)(i >> 7);
    const float tx = t * x, ty = t * y;
    const float txx = tx * x, tyy = ty * y, txy = tx * y;
#pragma unroll
    for (int l = 0; l < NLEV; ++l) {
      if (t >= thr[l]) {
        acc[6 * l + 0] += t;   acc[6 * l + 1] += tx;  acc[6 * l + 2] += ty;
        acc[6 * l + 3] += txx; acc[6 * l + 4] += tyy; acc[6 * l + 5] += txy;
      }
    }
  }
  block_sum<6 * NLEV>(acc, sred, sout);

  if (tid < NLEV) {
    float p[5];
    fit_params(&sout[6 * tid], p);
#pragma unroll
    for (int k = 0; k < 5; ++k) sparams[tid * 8 + k] = p[k];
    sparams[tid * 8 + 5] = cosf(p[2]);
    sparams[tid * 8 + 6] = sinf(p[2]);
  }
  __syncthreads();

  // --- metric pass: i_in/i_tot - area_frac per level ---
  float pcx[NLEV], pcy[NLEV], pco[NLEV], psi[NLEV], ia[NLEV], ib[NLEV];
#pragma unroll
  for (int l = 0; l < NLEV; ++l) {
    pcx[l] = sparams[l * 8 + 0]; pcy[l] = sparams[l * 8 + 1];
    pco[l] = sparams[l * 8 + 5]; psi[l] = sparams[l * 8 + 6];
    ia[l] = 1.0f / (sparams[l * 8 + 3] + EPSF);
    ib[l] = 1.0f / (sparams[l * 8 + 4] + EPSF);
  }
  float macc[2 * NLEV + 1];
#pragma unroll
  for (int k = 0; k < 2 * NLEV + 1; ++k) macc[k] = 0.f;

  for (int i = tid; i < NPIX; i += NT) {
    const float t = img[i];
    const float x = (float)(i & (W_ - 1));
    const float y = (float)(i >> 7);
    macc[2 * NLEV] += t;
#pragma unroll
    for (int l = 0; l < NLEV; ++l) {
      const float dx = x - pcx[l], dy = y - pcy[l];
      const float u = (dx * pco[l] + dy * psi[l]) * ia[l];
      const float v = (dy * pco[l] - dx * psi[l]) * ib[l];
      if (u * u + v * v <= 1.0f) {
        macc[2 * l] += t;
        macc[2 * l + 1] += 1.0f;
      }
    }
  }
  block_sum<2 * NLEV + 1>(macc, sred, sout);

  if (tid == 0) {
    const float itot = sout[2 * NLEV] + EPSF;
    int best = 0;
    float bm = -3.402823466e38f;
    for (int l = 0; l < NLEV; ++l) {
      const float m = sout[2 * l] / itot - sout[2 * l + 1] * (1.0f / (float)NPIX);
      if (m > bm) { bm = m; best = l; }             // first-max, like jnp.argmax
    }
    sbesti = best;
    for (int k = 0; k < 5; ++k) sbestp[k] = sparams[best * 8 + k];
  }
  __syncthreads();
  const int best = sbesti;

  // ===================== phase 2: output image ==========================
  stage_image(img, outg + (size_t)blk * NPIX);

  float mn2, mx2;
  block_minmax(img, sred, sout, mn2, mx2);
  const float inv2 = 1.0f / (mx2 - mn2 + EPSF);
  const float thr2 = c_lev[best] * ((mx2 - mn2) * inv2);

  float acc2[6] = {0.f, 0.f, 0.f, 0.f, 0.f, 0.f};
  for (int i = tid; i < NPIX; i += NT) {
    const float t = (img[i] - mn2) * inv2;
    if (t >= thr2) {
      const float x = (float)(i & (W_ - 1));
      const float y = (float)(i >> 7);
      const float tx = t * x, ty = t * y;
      acc2[0] += t;      acc2[1] += tx;     acc2[2] += ty;
      acc2[3] += tx * x; acc2[4] += ty * y; acc2[5] += tx * y;
    }
  }
  block_sum<6>(acc2, sred, sout);

  if (tid == 0) {
    float po[5];
    fit_params(sout, po);
    const float cx_t = sbestp[0], cy_t = sbestp[1], th_t = sbestp[2];
    const float a_t = sbestp[3], b_t = sbestp[4];
    const float scale = fmaxf(fmaxf(po[3], po[4]), fmaxf(a_t, b_t)) + EPSF;
    const float dcx = (po[0] - cx_t) / scale, dcy = (po[1] - cy_t) / scale;
    const float center = 0.5f * (dcx * dcx + dcy * dcy);
    const float cos_sim = cosf(po[2]) * cosf(th_t) + sinf(po[2]) * sinf(th_t);
    const float angle = 1.0f - cos_sim;
    const float da = (po[3] - a_t) / scale, db = (po[4] - b_t) / scale;
    const float axis = 0.5f * (da * da + db * db);
    loss[blk] = center + angle + axis;              // CW = AW = XW = 1
  }
}

// ------------------- WMMA mean reduction (512 -> 1) -------------------------
// A = data chunk (16x4 f32), B = ones => D[m,n] = rowsum(m); chain C across
// chunks. Rows partition the values, so sum of the 16 rows = grand total.

typedef __attribute__((ext_vector_type(2))) float v2f;
typedef __attribute__((ext_vector_type(8))) float v8f;

__global__ __launch_bounds__(32) void wmma_mean_kernel(
    const float* __restrict__ vals, float* __restrict__ outp, int n) {
  const int lane = threadIdx.x;
  // A[m,k]: lane m holds k=0,1 ; lane m+16 holds k=2,3  (16x4 f32 layout)
  const int base = 4 * (lane & 15) + 2 * (lane >> 4);
  v8f acc = {0.f, 0.f, 0.f, 0.f, 0.f, 0.f, 0.f, 0.f};
  v2f ones; ones.x = 1.0f; ones.y = 1.0f;
  const int chunks = n >> 6;                        // 64 values per WMMA
  for (int c = 0; c < chunks; ++c) {
    v2f a;
    a.x = vals[(c << 6) + base];
    a.y = vals[(c << 6) + base + 1];
    acc = __builtin_amdgcn_wmma_f32_16x16x4_f32(false, a, false, ones,
                                                (short)0, acc, false, false);
  }
  // lanes 0-15 hold rows 0-7 in acc[0..7]; lanes 16-31 hold rows 8-15.
  float s = acc[0] + acc[1] + acc[2] + acc[3] + acc[4] + acc[5] + acc[6] + acc[7];
  s += __shfl_xor(s, 16, 32);                       // grand total in every lane
  if (lane == 0) {
    for (int i = chunks << 6; i < n; ++i) s += vals[i];  // tail (unused for 512)
    outp[0] = s / (float)n;
  }
}

// ------------------------------- launcher -----------------------------------

extern "C" void kernel_launch(void* const* d_in, const int* in_sizes, int n_in,
                              void* d_out, int out_size, void* d_ws, size_t ws_size,
                              hipStream_t stream) {
  (void)n_in; (void)out_size; (void)ws_size;
  const float* outp = (const float*)d_in[0];   // "output"
  const float* tgtp = (const float*)d_in[1];   // "target"
  const int B = in_sizes[0] / NPIX;            // 512
  float* loss = (float*)d_ws;                  // B floats of scratch

  ellipse_best_loss_kernel<<<B, NT, 0, stream>>>(outp, tgtp, loss);
  wmma_mean_kernel<<<1, 32, 0, stream>>>(loss, (float*)d_out, B);
}